// ASMGEncoder_15642270892330
// MI455X (gfx1250) — compile-verified
//
#include <hip/hip_runtime.h>
#include <hip/hip_bf16.h>

#define HID 64

typedef __attribute__((ext_vector_type(2))) float v2f;
typedef __attribute__((ext_vector_type(8))) float v8f;

// ---------------------------------------------------------------------------
// zero fill (float4 vectorized)
// ---------------------------------------------------------------------------
__global__ __launch_bounds__(256) void zero_f4(float4* __restrict__ p, int n4) {
    int i = blockIdx.x * 256 + threadIdx.x;
    if (i < n4) p[i] = make_float4(0.f, 0.f, 0.f, 0.f);
}

// ---------------------------------------------------------------------------
// Y[r][n] = sum_k X[r][k]*W[n][k] + bias[n]     X:[rows,64] W:[64,64] Y:[rows,64]
// fp32 WMMA 16x16x4. Each wave owns one 16x16 output tile; 8 waves/block
// cover 32 rows x 64 cols. K-loop: 16 x (K=4) WMMA issues, branch-free:
// the ragged last row-tile clamps its A row index (duplicated valid row),
// and only the store is guarded -- OOB rows never reach memory.
// A frag (16x4): lane L -> row M=L%16, v[0]=K(lo of half), v[1]=K+1
// B frag (4x16): lane L -> col N=L%16 (B[k][n] = W[n][k]), same K split
// C/D (16x16):  VGPR p -> M = p + (L<16?0:8), N = L%16
// ---------------------------------------------------------------------------
__global__ __launch_bounds__(256) void gemm64_xwt_bias(
    const float* __restrict__ X, const float* __restrict__ W,
    const float* __restrict__ bias, float* __restrict__ Y, int rows)
{
    const int lane = threadIdx.x & 31;
    const int wave = threadIdx.x >> 5;
    const int m0 = (blockIdx.x * 2 + (wave >> 2)) * 16;
    const int n0 = (wave & 3) * 16;
    const int lm = lane & 15;
    const int ko = (lane < 16) ? 0 : 2;
    const int mrow = m0 + lm;
    const int mclamp = (mrow < rows) ? mrow : (rows - 1);   // branch-free A
    const int ncol = n0 + lm;

    const float* __restrict__ xr = X + (size_t)mclamp * HID + ko;
    const float* __restrict__ wr = W + (size_t)ncol * HID + ko;

    v8f c = {};
#pragma unroll
    for (int kk = 0; kk < 16; ++kk) {
        v2f a, b;
        a.x = xr[kk * 4];
        a.y = xr[kk * 4 + 1];
        b.x = wr[kk * 4];
        b.y = wr[kk * 4 + 1];
        c = __builtin_amdgcn_wmma_f32_16x16x4_f32(false, a, false, b,
                                                  (short)0, c, false, false);
    }
    const float bv = bias[ncol];
    const int mb = m0 + ((lane < 16) ? 0 : 8);
#pragma unroll
    for (int p = 0; p < 8; ++p) {
        int mr = mb + p;
        if (mr < rows) Y[(size_t)mr * HID + ncol] = c[p] + bv;
    }
}

// ---------------------------------------------------------------------------
// COO SpMM accumulate: out[r,:] += v * x[c,:]   (H=64)
// One wave per edge; lane handles a float2 of the row. f32 atomics hit L2
// (state matrices are 25 MB << 192 MB L2). Adjacency selector is read from
// device memory (idxes_* live on device; no host readback allowed).
// ---------------------------------------------------------------------------
__global__ __launch_bounds__(256) void spmm_coo_atomic(
    const int* __restrict__ rows_all, const int* __restrict__ cols_all,
    const float* __restrict__ vals_all, const int* __restrict__ aidx_ptr,
    int nnz, const float* __restrict__ x, float* __restrict__ out)
{
    const int a = *aidx_ptr;
    const size_t base = (size_t)a * (size_t)nnz;
    const int e = blockIdx.x * 8 + (threadIdx.x >> 5);
    if (e >= nnz) return;
    const int lane = threadIdx.x & 31;

    if (e + 4096 < nnz) {  // stream-ahead on the COO arrays (global_prefetch_b8)
        __builtin_prefetch(&rows_all[base + e + 4096], 0, 0);
        __builtin_prefetch(&cols_all[base + e + 4096], 0, 0);
    }

    const int   r = rows_all[base + e];
    const int   ci = cols_all[base + e];
    const float v = vals_all[base + e];

    const float2 xv = ((const float2*)(x + (size_t)ci * HID))[lane];
    float* op = out + (size_t)r * HID + (size_t)lane * 2;
    unsafeAtomicAdd(op,     v * xv.x);   // global_atomic_add_f32
    unsafeAtomicAdd(op + 1, v * xv.y);
}

// ---------------------------------------------------------------------------
// Fused: LayerNorm -> exact GELU -> store temp, and attention logit
//   logit = tanh(h @ A1^T + b1) @ A2^T + b2
// One wave per node (wave32): shuffle reductions for mean/var/logit,
// LDS broadcast of the 64-vector for the 64x64 matvec.
// ---------------------------------------------------------------------------
__device__ inline float wave_sum(float v) {
#pragma unroll
    for (int o = 16; o > 0; o >>= 1) v += __shfl_xor(v, o, 32);
    return v;
}

__global__ __launch_bounds__(256) void ln_gelu_attn(
    const float* __restrict__ s, const float* __restrict__ g,
    const float* __restrict__ b, const float* __restrict__ a1W,
    const float* __restrict__ a1b, const float* __restrict__ a2W,
    const float* __restrict__ a2b, float* __restrict__ temp,
    float* __restrict__ attn, int n, int m)
{
    __shared__ float sh[8][HID];
    const int lane = threadIdx.x & 31;
    const int wv = threadIdx.x >> 5;
    const int nodeRaw = blockIdx.x * 8 + wv;
    const bool valid = nodeRaw < n;
    const int node = valid ? nodeRaw : (n - 1);

    float2 hv = ((const float2*)(s + (size_t)node * HID))[lane];
    const float mu = wave_sum(hv.x + hv.y) * (1.f / HID);
    const float dx = hv.x - mu, dy = hv.y - mu;
    const float var = wave_sum(dx * dx + dy * dy) * (1.f / HID);
    const float rstd = rsqrtf(var + 1e-5f);
    float h0 = dx * rstd * g[lane * 2]     + b[lane * 2];
    float h1 = dy * rstd * g[lane * 2 + 1] + b[lane * 2 + 1];
    // exact gelu: 0.5*x*(1+erf(x/sqrt(2)))
    h0 = 0.5f * h0 * (1.f + erff(h0 * 0.7071067811865475f));
    h1 = 0.5f * h1 * (1.f + erff(h1 * 0.7071067811865475f));

    sh[wv][lane * 2] = h0;
    sh[wv][lane * 2 + 1] = h1;
    __syncthreads();

    float acc0 = a1b[lane * 2], acc1 = a1b[lane * 2 + 1];
    const float* w0 = a1W + (size_t)(lane * 2) * HID;
    const float* w1 = w0 + HID;
#pragma unroll
    for (int k = 0; k < HID; ++k) {
        const float hk = sh[wv][k];
        acc0 += hk * w0[k];
        acc1 += hk * w1[k];
    }
    const float t0 = tanhf(acc0), t1 = tanhf(acc1);
    const float logit = wave_sum(t0 * a2W[lane * 2] + t1 * a2W[lane * 2 + 1]);

    if (valid) {
        ((float2*)(temp + (size_t)node * HID))[lane] = make_float2(h0, h1);
        if (lane == 0) attn[(size_t)node * 2 + m] = logit + a2b[0];
    }
}

// ---------------------------------------------------------------------------
// out[n,:] = softmax(attn[n,:]) . {temp0[n,:], temp1[n,:]}   (M = 2)
// ---------------------------------------------------------------------------
__global__ __launch_bounds__(256) void softmax_combine(
    const float* __restrict__ t0, const float* __restrict__ t1,
    const float* __restrict__ attn, float* __restrict__ out, int n)
{
    const int tid = blockIdx.x * 256 + threadIdx.x;
    if (tid >= n * 32) return;
    const int node = tid >> 5, lane = tid & 31;
    const float a0 = attn[(size_t)node * 2];
    const float a1 = attn[(size_t)node * 2 + 1];
    const float mx = fmaxf(a0, a1);
    const float e0 = expf(a0 - mx), e1 = expf(a1 - mx);
    const float inv = 1.f / (e0 + e1);
    const float p0 = e0 * inv, p1 = e1 * inv;
    const float2 v0 = ((const float2*)(t0 + (size_t)node * HID))[lane];
    const float2 v1 = ((const float2*)(t1 + (size_t)node * HID))[lane];
    ((float2*)(out + (size_t)node * HID))[lane] =
        make_float2(p0 * v0.x + p1 * v1.x, p0 * v0.y + p1 * v1.y);
}

// ---------------------------------------------------------------------------
extern "C" void kernel_launch(void* const* d_in, const int* in_sizes, int n_in,
                              void* d_out, int out_size, void* d_ws, size_t ws_size,
                              hipStream_t stream)
{
    const float* feats[3] = {(const float*)d_in[0], (const float*)d_in[1],
                             (const float*)d_in[2]};
    const float* ws_W = (const float*)d_in[3];
    const float* ws_b = (const float*)d_in[4];
    const float* adj_vals = (const float*)d_in[5];
    const float* aff_W = (const float*)d_in[6];
    const float* aff_b = (const float*)d_in[7];
    const float* ng  = (const float*)d_in[8];
    const float* nbp = (const float*)d_in[9];
    const float* a1W = (const float*)d_in[10];
    const float* a1b = (const float*)d_in[11];
    const float* a2W = (const float*)d_in[12];
    const float* a2b = (const float*)d_in[13];
    const int* adj_rows = (const int*)d_in[15];
    const int* adj_cols = (const int*)d_in[16];
    const int* idx_seq = (const int*)d_in[17];
    const int* idx_res = (const int*)d_in[18];

    const int npt = in_sizes[0] / HID;   // nodes per type (33000)
    const int n = 3 * npt;               // 99000
    const int nnz = in_sizes[5] / 6;     // 1,584,000

    const size_t nh = (size_t)n * HID;
    float* wsf = (float*)d_ws;
    float* hid = wsf;
    float* st0 = wsf + 1 * nh;
    float* st1 = wsf + 2 * nh;
    float* st2 = wsf + 3 * nh;
    float* st3 = wsf + 4 * nh;
    float* tmp[2] = {wsf + 5 * nh, wsf + 6 * nh};
    float* attn = wsf + 7 * nh;

    const dim3 blk(256);
    const dim3 pgrid((npt + 31) / 32);
    const dim3 fgrid((n + 31) / 32);
    const int nh4 = (int)(nh / 4);
    const dim3 zgrid((nh4 + 255) / 256);
    const dim3 sgrid((nnz + 7) / 8);
    const dim3 agrid((n + 7) / 8);
    const dim3 cgrid((n * 32 + 255) / 256);

    // type-wise input projection into contiguous hid blocks
    for (int t = 0; t < 3; ++t)
        gemm64_xwt_bias<<<pgrid, blk, 0, stream>>>(
            feats[t], ws_W + t * HID * HID, ws_b + t * HID,
            hid + (size_t)t * npt * HID, npt);

    for (int m = 0; m < 2; ++m) {
        // x = hid @ cell_aff_W[m]^T + cell_aff_b[m]
        gemm64_xwt_bias<<<fgrid, blk, 0, stream>>>(
            hid, aff_W + m * HID * HID, aff_b + m * HID, st0, n);

        // states[1] = spmm(seq0, s0)
        zero_f4<<<zgrid, blk, 0, stream>>>((float4*)st1, nh4);
        spmm_coo_atomic<<<sgrid, blk, 0, stream>>>(adj_rows, adj_cols, adj_vals,
                                                   idx_seq + m * 3 + 0, nnz, st0, st1);
        // states[2] = spmm(seq1, s1) + spmm(res0, s0)
        zero_f4<<<zgrid, blk, 0, stream>>>((float4*)st2, nh4);
        spmm_coo_atomic<<<sgrid, blk, 0, stream>>>(adj_rows, adj_cols, adj_vals,
                                                   idx_seq + m * 3 + 1, nnz, st1, st2);
        spmm_coo_atomic<<<sgrid, blk, 0, stream>>>(adj_rows, adj_cols, adj_vals,
                                                   idx_res + m * 3 + 0, nnz, st0, st2);
        // states[3] = spmm(seq2, s2) + spmm(res1, s0) + spmm(res2, s1)
        zero_f4<<<zgrid, blk, 0, stream>>>((float4*)st3, nh4);
        spmm_coo_atomic<<<sgrid, blk, 0, stream>>>(adj_rows, adj_cols, adj_vals,
                                                   idx_seq + m * 3 + 2, nnz, st2, st3);
        spmm_coo_atomic<<<sgrid, blk, 0, stream>>>(adj_rows, adj_cols, adj_vals,
                                                   idx_res + m * 3 + 1, nnz, st0, st3);
        spmm_coo_atomic<<<sgrid, blk, 0, stream>>>(adj_rows, adj_cols, adj_vals,
                                                   idx_res + m * 3 + 2, nnz, st1, st3);

        // layernorm + gelu + attention logit
        ln_gelu_attn<<<agrid, blk, 0, stream>>>(st3, ng + m * HID, nbp + m * HID,
                                                a1W, a1b, a2W, a2b,
                                                tmp[m], attn, n, m);
    }

    softmax_combine<<<cgrid, blk, 0, stream>>>(tmp[0], tmp[1], attn,
                                               (float*)d_out, n);
}